// GCN_65979287601497
// MI455X (gfx1250) — compile-verified
//
#include <hip/hip_runtime.h>
#include <hip/hip_bf16.h>

typedef __attribute__((ext_vector_type(2))) float v2f;
typedef __attribute__((ext_vector_type(8))) float v8f;
typedef __attribute__((ext_vector_type(4))) unsigned int v4u;
typedef __attribute__((ext_vector_type(4))) int v4i;
typedef __attribute__((ext_vector_type(8))) int v8i;

// ---------------------------------------------------------------------------
// Degree / normalization
// ---------------------------------------------------------------------------
__global__ void deg_init_kernel(int* __restrict__ degi, int N) {
    int i = blockIdx.x * blockDim.x + threadIdx.x;
    if (i < N) degi[i] = 0;
}

__global__ void deg_count_kernel(const int* __restrict__ col, int* __restrict__ degi, int E) {
    int e = blockIdx.x * blockDim.x + threadIdx.x;
    if (e < E) atomicAdd(&degi[col[e]], 1);
}

__global__ void dis_kernel(const int* __restrict__ degi, float* __restrict__ dis, int N) {
    int i = blockIdx.x * blockDim.x + threadIdx.x;
    if (i < N) dis[i] = rsqrtf((float)degi[i] + 1.0f);
}

__global__ void norm_kernel(const int* __restrict__ row, const int* __restrict__ col,
                            const float* __restrict__ dis, float* __restrict__ nrm, int E) {
    int e = blockIdx.x * blockDim.x + threadIdx.x;
    if (e < E) nrm[e] = dis[row[e]] * dis[col[e]];
}

// ---------------------------------------------------------------------------
// Embedding gather + sum over K features: h[n][:] = sum_k emb[x_idx[n][k]][:]
// ---------------------------------------------------------------------------
__global__ void embed_sum_kernel(const int* __restrict__ xidx, const float* __restrict__ emb,
                                 float* __restrict__ h, int N, int Kf, int D) {
    int nd4 = D >> 2;
    int idx = blockIdx.x * blockDim.x + threadIdx.x;
    if (idx >= N * nd4) return;
    int n = idx / nd4, d4 = idx % nd4;
    float4 s = make_float4(0.f, 0.f, 0.f, 0.f);
    for (int k = 0; k < Kf; ++k) {
        int v = xidx[n * Kf + k];
        float4 e = ((const float4*)(emb + (size_t)v * D))[d4];
        s.x += e.x; s.y += e.y; s.z += e.z; s.w += e.w;
    }
    ((float4*)(h + (size_t)n * D))[d4] = s;
}

// ---------------------------------------------------------------------------
// TDM helper: DMA a 2D tile (tile1 rows x tile0 elems, fp32) from global into
// LDS with hardware row padding (1 DWORD pad every 2^padIntervalLog DWORDs,
// giving odd LDS pitch for bank-conflict-free fragment reads).
// D# packing per CDNA5 ISA 8.3 / 8.4:
//   group0: count=1 | lds_addr | global_addr[56:0] | type=2
//   group1: data_size=2(4B), pad_enable=1, pad_interval, pad_amount=0(1 DWORD),
//           tensor_dim0/1 (OOB rows/cols read as zero), tile_dim0/1,
//           tensor_dim0_stride; dims 2+ unused (groups 2/3 zero).
// ---------------------------------------------------------------------------
__device__ __forceinline__ void tdm_load_tile_2d(
        unsigned ldsOff, const float* gptr,
        unsigned tensorD0, unsigned tensorD1,
        unsigned tile0, unsigned tile1,
        unsigned strideElems, unsigned padIntervalLog)
{
    unsigned long long ga = (unsigned long long)(size_t)gptr;
    v4u g0;
    g0.x = 1u;                                                  // count=1 valid user D#
    g0.y = ldsOff;                                              // LDS byte address
    g0.z = (unsigned)ga;                                        // global_addr[31:0]
    g0.w = (unsigned)((ga >> 32) & 0x01FFFFFFu) | (2u << 30);   // addr[56:32] | type=2

    // pad_interval encoding: 0=2,1=4,2=8,3=16,4=32,5=64,... DWORDs
    unsigned padIv = padIntervalLog - 1u;
    v8i g1;
    g1[0] = (int)((2u << 16) | (1u << 20) | (padIv << 22));     // data_size=4B, pad_en, interval
    g1[1] = (int)((tensorD0 & 0xFFFFu) << 16);                  // tensor_dim0[15:0]
    g1[2] = (int)(((tensorD0 >> 16) & 0xFFFFu) | ((tensorD1 & 0xFFFFu) << 16));
    g1[3] = (int)(((tensorD1 >> 16) & 0xFFFFu) | ((tile0 & 0xFFFFu) << 16));
    g1[4] = (int)(tile1 & 0xFFFFu);                             // tile_dim1, tile_dim2=0
    g1[5] = (int)strideElems;                                   // tensor_dim0_stride[31:0]
    g1[6] = 0;                                                  // stride[47:32]=0, dim1_stride lo
    g1[7] = 0;

    v4i z4 = {0, 0, 0, 0};
#if defined(__clang_major__) && (__clang_major__ >= 23)
    v8i z8 = {0, 0, 0, 0, 0, 0, 0, 0};
    __builtin_amdgcn_tensor_load_to_lds(g0, g1, z4, z4, z8, 0);
#else
    __builtin_amdgcn_tensor_load_to_lds(g0, g1, z4, z4, 0);
#endif
}

// ---------------------------------------------------------------------------
// WMMA fp32 GEMM: C[M x Nout] = A[M x K] @ W[K x Nout]
// Block tile 64(M) x 64(N), 8 waves; each wave computes 16x32 (two 16x16
// accumulators sharing one A fragment) with V_WMMA_F32_16X16X4_F32.
// Tiles staged into double-buffered LDS by the Tensor Data Mover (wave 0 -> A,
// wave 1 -> B), overlapped with compute; s_wait_tensorcnt + barrier sync.
// LDS pitches: A 33 floats/row, B 65 floats/row (TDM pad insertion).
// ---------------------------------------------------------------------------
#define A_PITCH 33
#define B_PITCH 65

__global__ void __launch_bounds__(256) gemm_wmma_f32_kernel(
        const float* __restrict__ A, const float* __restrict__ W,
        float* __restrict__ C, int M, int K, int Nout)
{
    __shared__ float As[2][64 * A_PITCH];
    __shared__ float Bs[2][32 * B_PITCH];

    const int tid  = threadIdx.x;
    const int lane = tid & 31;
    const int wave = tid >> 5;
    const int l16  = lane & 15;
    const int half = lane >> 4;          // 0: lanes 0-15, 1: lanes 16-31
    const int m0   = (wave & 3) * 16;    // wave's M sub-tile within block
    const int n0   = (wave >> 2) * 32;   // wave's N sub-tile within block
    const int rowBase = blockIdx.x * 64;
    const int colBase = blockIdx.y * 64;

    v8f acc0 = {0.f, 0.f, 0.f, 0.f, 0.f, 0.f, 0.f, 0.f};
    v8f acc1 = {0.f, 0.f, 0.f, 0.f, 0.f, 0.f, 0.f, 0.f};

    // Prologue: DMA first K-chunk into buffer 0.
    if (wave == 0)
        tdm_load_tile_2d((unsigned)(size_t)&As[0][0], A + (size_t)rowBase * K,
                         (unsigned)K, (unsigned)(M - rowBase), 32u, 64u, (unsigned)K, 5u);
    if (wave == 1)
        tdm_load_tile_2d((unsigned)(size_t)&Bs[0][0], W + colBase,
                         (unsigned)(Nout - colBase), (unsigned)K, 64u, 32u, (unsigned)Nout, 6u);

    int buf = 0;
    for (int k0 = 0; k0 < K; k0 += 32) {
        if (wave < 2) __builtin_amdgcn_s_wait_tensorcnt(0);
        __syncthreads();                           // tiles for k0 visible to all waves

        const int nbuf = buf ^ 1;
        if (k0 + 32 < K) {                         // prefetch next chunk into other buffer
            if (wave == 0)
                tdm_load_tile_2d((unsigned)(size_t)&As[nbuf][0],
                                 A + (size_t)rowBase * K + (k0 + 32),
                                 (unsigned)(K - (k0 + 32)), (unsigned)(M - rowBase),
                                 32u, 64u, (unsigned)K, 5u);
            if (wave == 1)
                tdm_load_tile_2d((unsigned)(size_t)&Bs[nbuf][0],
                                 W + (size_t)(k0 + 32) * Nout + colBase,
                                 (unsigned)(Nout - colBase), (unsigned)(K - (k0 + 32)),
                                 64u, 32u, (unsigned)Nout, 6u);
        }

        const float* as = &As[buf][0];
        const float* bs = &Bs[buf][0];
#pragma unroll
        for (int kk = 0; kk < 32; kk += 4) {
            // A 16x4 fragment: lanes 0-15 K={kk,kk+1}, lanes 16-31 K={kk+2,kk+3}
            v2f a, b0, b1;
            const int ar = (m0 + l16) * A_PITCH + kk + 2 * half;
            a.x = as[ar];
            a.y = as[ar + 1];
            // B 4x16 fragments (two N sub-tiles share the A fragment)
            const int br = (kk + 2 * half) * B_PITCH + l16;
            b0.x = bs[br + n0];
            b0.y = bs[br + B_PITCH + n0];
            b1.x = bs[br + n0 + 16];
            b1.y = bs[br + B_PITCH + n0 + 16];
            acc0 = __builtin_amdgcn_wmma_f32_16x16x4_f32(
                false, a, false, b0, (short)0, acc0, false, false);
            acc1 = __builtin_amdgcn_wmma_f32_16x16x4_f32(
                false, a, false, b1, (short)0, acc1, false, false);
        }
        __syncthreads();                           // done reading buf before it is reused
        buf = nbuf;
    }

    // C/D layout: VGPR i -> M = m0 + i + 8*half, N = n0 + l16 (and +16)
#pragma unroll
    for (int i = 0; i < 8; ++i) {
        int gr = rowBase + m0 + i + 8 * half;
        if (gr < M) {
            C[(size_t)gr * Nout + (colBase + n0 + l16)]      = acc0[i];
            C[(size_t)gr * Nout + (colBase + n0 + 16 + l16)] = acc1[i];
        }
    }
}

// ---------------------------------------------------------------------------
// agg[n][f] = t[n][f] * dis[n]^2 + bias[f]   (self-loop term + bias init)
// ---------------------------------------------------------------------------
__global__ void agg_init_kernel(const float* __restrict__ t, const float* __restrict__ dis,
                                const float* __restrict__ bias, float* __restrict__ agg,
                                int N, int F) {
    int idx = blockIdx.x * blockDim.x + threadIdx.x;
    if (idx >= N * F) return;
    int n = idx / F, f = idx % F;
    float d = dis[n];
    agg[idx] = t[idx] * d * d + bias[f];
}

// ---------------------------------------------------------------------------
// Edge scatter: agg[col[e]][:] += t[row[e]][:] * norm[e]
// One block per edge; blockDim = F/4 threads, float4 source reads.
// ---------------------------------------------------------------------------
__global__ void scatter_kernel(const float* __restrict__ t, const int* __restrict__ row,
                               const int* __restrict__ col, const float* __restrict__ nrm,
                               float* __restrict__ agg, int F) {
    int e = blockIdx.x;
    int r = row[e], c = col[e];
    float w = nrm[e];
    int f = threadIdx.x << 2;
    float4 v = *(const float4*)(t + (size_t)r * F + f);
    float* dst = agg + (size_t)c * F + f;
    atomicAdd(dst + 0, v.x * w);
    atomicAdd(dst + 1, v.y * w);
    atomicAdd(dst + 2, v.z * w);
    atomicAdd(dst + 3, v.w * w);
}

__global__ void relu_copy_kernel(const float* __restrict__ src, float* __restrict__ dst, int n) {
    int i = blockIdx.x * blockDim.x + threadIdx.x;
    if (i < n) dst[i] = fmaxf(src[i], 0.f);
}

__global__ void residual_relu_kernel(float* __restrict__ h, const float* __restrict__ agg, int n) {
    int i = blockIdx.x * blockDim.x + threadIdx.x;
    if (i < n) h[i] = h[i] + fmaxf(agg[i], 0.f);
}

// ---------------------------------------------------------------------------
// Host-side orchestration
// ---------------------------------------------------------------------------
extern "C" void kernel_launch(void* const* d_in, const int* in_sizes, int n_in,
                              void* d_out, int out_size, void* d_ws, size_t ws_size,
                              hipStream_t stream) {
    const int*   x_idx = (const int*)d_in[0];
    const int*   eidx  = (const int*)d_in[1];
    const float* emb   = (const float*)d_in[2];
    const float* W0    = (const float*)d_in[3];
    const float* b0    = (const float*)d_in[4];
    const float* W1    = (const float*)d_in[5];
    const float* b1    = (const float*)d_in[6];
    const float* W2    = (const float*)d_in[7];
    const float* b2    = (const float*)d_in[8];
    float* out = (float*)d_out;

    // Derive problem sizes from input shapes.
    const int H  = in_sizes[4];            // b0: [H] = 384
    const int D  = in_sizes[8];            // b2: [D] = 768
    const int N  = out_size / D;           // output: [N, D]
    const int Kf = in_sizes[0] / N;        // x_idx: [N, Kf]
    const int E  = in_sizes[1] / 2;        // edge_index: [2, E]
    const int* row = eidx;
    const int* col = eidx + E;

    // Workspace carve-up (fp32 units).
    float* ws = (float*)d_ws;
    size_t o = 0;
    int*   degi = (int*)(ws + o); o += (size_t)N;
    float* dis  = ws + o;         o += (size_t)N;
    float* nrm  = ws + o;         o += (size_t)E;
    float* h0   = ws + o;         o += (size_t)N * D;   // embed out; reused for layer-2 GEMM out
    float* hA   = ws + o;         o += (size_t)N * H;   // layer activations (residual carrier)
    float* tB   = ws + o;         o += (size_t)N * H;   // GEMM output (layers 0,1)
    float* aggB = ws + o;         o += (size_t)N * H;   // aggregation buffer (layers 0,1)
    (void)ws_size; (void)n_in;

    const int T = 256;
    auto cdiv = [](int a, int b) { return (a + b - 1) / b; };

    // --- degree, dis, per-edge norm ---
    deg_init_kernel<<<cdiv(N, T), T, 0, stream>>>(degi, N);
    deg_count_kernel<<<cdiv(E, T), T, 0, stream>>>(col, degi, E);
    dis_kernel<<<cdiv(N, T), T, 0, stream>>>(degi, dis, N);
    norm_kernel<<<cdiv(E, T), T, 0, stream>>>(row, col, dis, nrm, E);

    // --- embedding gather+sum: h0 = emb[x_idx].sum(1)  [N, D] ---
    embed_sum_kernel<<<cdiv(N * (D / 4), T), T, 0, stream>>>(x_idx, emb, h0, N, Kf, D);

    const int mBlocks = cdiv(N, 64);

    // --- Layer 0: t = h0 @ W0 ; agg ; hA = relu(agg) ---
    gemm_wmma_f32_kernel<<<dim3(mBlocks, H / 64), T, 0, stream>>>(h0, W0, tB, N, D, H);
    agg_init_kernel<<<cdiv(N * H, T), T, 0, stream>>>(tB, dis, b0, aggB, N, H);
    scatter_kernel<<<E, H / 4, 0, stream>>>(tB, row, col, nrm, aggB, H);
    relu_copy_kernel<<<cdiv(N * H, T), T, 0, stream>>>(aggB, hA, N * H);

    // --- Layer 1: t = hA @ W1 ; agg ; hA += relu(agg) ---
    gemm_wmma_f32_kernel<<<dim3(mBlocks, H / 64), T, 0, stream>>>(hA, W1, tB, N, H, H);
    agg_init_kernel<<<cdiv(N * H, T), T, 0, stream>>>(tB, dis, b1, aggB, N, H);
    scatter_kernel<<<E, H / 4, 0, stream>>>(tB, row, col, nrm, aggB, H);
    residual_relu_kernel<<<cdiv(N * H, T), T, 0, stream>>>(hA, aggB, N * H);

    // --- Layer 2: t2 = hA @ W2 (reuse h0 buffer) ; aggregate straight into out ---
    gemm_wmma_f32_kernel<<<dim3(mBlocks, D / 64), T, 0, stream>>>(hA, W2, h0, N, H, D);
    agg_init_kernel<<<cdiv(N * D, T), T, 0, stream>>>(h0, dis, b2, out, N, D);
    scatter_kernel<<<E, D / 4, 0, stream>>>(h0, row, col, nrm, out, D);
}